// GraphConvolution_86517821212326
// MI455X (gfx1250) — compile-verified
//
#include <hip/hip_runtime.h>
#include <hip/hip_bf16.h>

// Problem constants (from the reference): N=50000, K=32, DIN=256, DOUT=128
#define DIN   256
#define DOUT  128
#define KNBR  32
#define FROW  260               // padded LDS row stride (floats): 260 % 64banks = 4 -> conflict-free
#define TILE_M 16

typedef float v2f __attribute__((ext_vector_type(2)));
typedef float v8f __attribute__((ext_vector_type(8)));

// -------------------------------------------------------------------------
// Kernel 1: h = relu(feats @ W^T + b) using V_WMMA_F32_16X16X4_F32.
// Block = 256 threads = 8 waves. Whole W (128x256 f32) staged in LDS once
// per block (CDNA5 WGP has 320KB LDS). Grid-stride over 16-row tiles of
// feats; each wave owns one 16-column tile of the output.
// -------------------------------------------------------------------------
__global__ __launch_bounds__(256) void fc_relu_wmma(
    const float* __restrict__ feats,   // [N, DIN]
    const float* __restrict__ Wm,      // [DOUT, DIN]
    const float* __restrict__ bias,    // [DOUT]
    float* __restrict__ h,             // [N, DOUT]  (workspace)
    int N)
{
    extern __shared__ float smem[];
    float* Wl = smem;                     // [DOUT][FROW]
    float* Fl = smem + DOUT * FROW;       // [TILE_M][FROW]

    const int tid  = threadIdx.x;
    const int wave = tid >> 5;            // 0..7 -> column tile
    const int lane = tid & 31;

    // ---- Stage whole W into LDS (row-padded), float4 loads/stores ----
    for (int i = tid; i < (DOUT * DIN) / 4; i += 256) {
        const int r = i >> 6;             // 64 float4 per row
        const int c = (i & 63) << 2;
        const float4 v = reinterpret_cast<const float4*>(Wm)[i];
        *reinterpret_cast<float4*>(&Wl[r * FROW + c]) = v;
    }
    __syncthreads();

    // Per-lane WMMA fragment coordinates (ISA 7.12.2, 32-bit 16x4 A layout):
    //   lanes 0-15 : M/N = lane,    K-pair {k, k+1}
    //   lanes 16-31: M/N = lane-16, K-pair {k+2, k+3}
    const int m  = lane & 15;
    const int kb = (lane >> 4) << 1;          // 0 or 2
    const int col = wave * TILE_M + m;        // this wave's output column for B
    const float bcol = bias[col];
    const int mBase = (lane >> 4) << 3;       // D: VGPR i holds M=i (lo half) / M=i+8 (hi half)

    const float* wRow = &Wl[col * FROW + kb]; // B fragment source (invariant)

    const int nRowTiles = (N + TILE_M - 1) / TILE_M;
    for (int rt = blockIdx.x; rt < nRowTiles; rt += gridDim.x) {
        const int row0 = rt * TILE_M;

        // ---- Stage 16xDIN feats tile into LDS ----
        for (int i = tid; i < (TILE_M * DIN) / 4; i += 256) {
            const int r = i >> 6;
            const int c = (i & 63) << 2;
            const int gr = row0 + r;
            float4 v = make_float4(0.f, 0.f, 0.f, 0.f);
            if (gr < N)
                v = reinterpret_cast<const float4*>(feats)[gr * (DIN / 4) + (i & 63)];
            *reinterpret_cast<float4*>(&Fl[r * FROW + c]) = v;
        }
        __syncthreads();

        // ---- 64 chained f32 WMMAs over K = 256 ----
        const float* fRow = &Fl[m * FROW + kb];
        v8f acc = {};
#pragma unroll
        for (int k = 0; k < DIN; k += 4) {
            const v2f a  = *reinterpret_cast<const v2f*>(fRow + k);
            const v2f bb = *reinterpret_cast<const v2f*>(wRow + k);
            acc = __builtin_amdgcn_wmma_f32_16x16x4_f32(
                /*neg_a=*/false, a, /*neg_b=*/false, bb,
                /*c_mod=*/(short)0, acc, /*reuse_a=*/false, /*reuse_b=*/false);
        }

        // ---- bias + relu + store D (16x16 tile, documented C/D layout) ----
#pragma unroll
        for (int i = 0; i < 8; ++i) {
            const int gr = row0 + mBase + i;
            if (gr < N) {
                float v = acc[i] + bcol;
                h[gr * DOUT + col] = v > 0.f ? v : 0.f;
            }
        }
        __syncthreads();
    }
}

// -------------------------------------------------------------------------
// Kernel 2: out[n] = mean_j h[edge[n][j]]  — one wave per node.
// h (25.6MB) is L2-resident (192MB L2), each neighbor row read is a fully
// coalesced 512B burst (32 lanes x float4).
// -------------------------------------------------------------------------
__global__ __launch_bounds__(256) void gather_mean(
    const float* __restrict__ h,       // [N, DOUT]
    const int*   __restrict__ edges,   // [N, KNBR]
    float* __restrict__ out,           // [N, DOUT]
    int N)
{
    const int wave = threadIdx.x >> 5;
    const int lane = threadIdx.x & 31;
    const int node = blockIdx.x * 8 + wave;
    if (node >= N) return;

    const int myIdx = edges[node * KNBR + lane];  // lane j holds neighbor j

    float4 acc = make_float4(0.f, 0.f, 0.f, 0.f);
#pragma unroll 8
    for (int j = 0; j < KNBR; ++j) {
        const int idx = __shfl(myIdx, j, 32);
        const float4 v = reinterpret_cast<const float4*>(h)[idx * (DOUT / 4) + lane];
        acc.x += v.x; acc.y += v.y; acc.z += v.z; acc.w += v.w;
    }
    const float s = 1.0f / (float)KNBR;
    float4 r = make_float4(acc.x * s, acc.y * s, acc.z * s, acc.w * s);
    reinterpret_cast<float4*>(out)[node * (DOUT / 4) + lane] = r;
}

// -------------------------------------------------------------------------
extern "C" void kernel_launch(void* const* d_in, const int* in_sizes, int n_in,
                              void* d_out, int out_size, void* d_ws, size_t ws_size,
                              hipStream_t stream)
{
    // setup_inputs order: ids, feats, W, b, edge_dict, G, ite
    const float* feats = (const float*)d_in[1];
    const float* Wm    = (const float*)d_in[2];
    const float* bias  = (const float*)d_in[3];
    const int*   edges = (const int*)d_in[4];
    float*       out   = (float*)d_out;

    const int N = in_sizes[4] / KNBR;           // 50000
    float* h = (float*)d_ws;                     // N*DOUT floats of scratch

    const int nRowTiles = (N + TILE_M - 1) / TILE_M;
    int grid1 = nRowTiles < 512 ? nRowTiles : 512;       // persistent-ish: amortize W staging
    const size_t smemBytes = (size_t)(DOUT * FROW + TILE_M * FROW) * sizeof(float); // ~146KB of 320KB WGP LDS

    fc_relu_wmma<<<grid1, 256, smemBytes, stream>>>(feats, Wm, bias, h, N);

    const int grid2 = (N + 7) / 8;                        // one wave per node
    gather_mean<<<grid2, 256, 0, stream>>>(h, edges, out, N);
}